// ETracking_Net_86526411145613
// MI455X (gfx1250) — compile-verified
//
#include <hip/hip_runtime.h>
#include <hip/hip_bf16.h>
#include <math.h>

typedef float v2f __attribute__((ext_vector_type(2)));
typedef float v8f __attribute__((ext_vector_type(8)));

constexpr int Bc = 4;        // batch
constexpr int Cc = 5;        // channels
constexpr int Pc = 32768;    // points
constexpr int SMAX = 3136;   // num_points in setup
constexpr int STILES = (SMAX + 15) / 16;  // 196 sample tiles per batch

__device__ __forceinline__ v8f wmma16x16x4_f32(v2f a, v2f b, v8f c) {
#if defined(__AMDGCN__) && __has_builtin(__builtin_amdgcn_wmma_f32_16x16x4_f32)
  // (neg_a, A, neg_b, B, c_mod, C, reuse_a, reuse_b)
  return __builtin_amdgcn_wmma_f32_16x16x4_f32(false, a, false, b, (short)0, c,
                                               false, false);
#else
  return c;  // host pass / unsupported target: never executed on gfx1250
#endif
}

__device__ __forceinline__ float min8(v8f c) {
  return fminf(fminf(fminf(c[0], c[1]), fminf(c[2], c[3])),
               fminf(fminf(c[4], c[5]), fminf(c[6], c[7])));
}

// ---------------------------------------------------------------------------
// Kernel 1: per-batch min/max of channels 0,1 and precompute squared norms
// ---------------------------------------------------------------------------
__global__ void prep_kernel(const float* __restrict__ x, float* __restrict__ minmax,
                            float* __restrict__ xx2, float* __restrict__ xx3) {
  int b = blockIdx.x;
  const float* x0 = x + (size_t)b * Cc * Pc;
  const float* x1 = x0 + Pc;
  const float* x2 = x0 + 2 * Pc;
  float mn0 = INFINITY, mx0 = -INFINITY, mn1 = INFINITY, mx1 = -INFINITY;
  for (int p = threadIdx.x; p < Pc; p += blockDim.x) {
    float a = x0[p], c = x1[p], d = x2[p];
    mn0 = fminf(mn0, a); mx0 = fmaxf(mx0, a);
    mn1 = fminf(mn1, c); mx1 = fmaxf(mx1, c);
    float s2 = a * a + c * c;
    xx2[(size_t)b * Pc + p] = s2;
    xx3[(size_t)b * Pc + p] = s2 + d * d;
  }
  __shared__ float r0[256], r1[256], r2[256], r3[256];
  int t = threadIdx.x;
  r0[t] = mn0; r1[t] = mx0; r2[t] = mn1; r3[t] = mx1;
  __syncthreads();
  for (int st = 128; st > 0; st >>= 1) {
    if (t < st) {
      r0[t] = fminf(r0[t], r0[t + st]); r1[t] = fmaxf(r1[t], r1[t + st]);
      r2[t] = fminf(r2[t], r2[t + st]); r3[t] = fmaxf(r3[t], r3[t + st]);
    }
    __syncthreads();
  }
  if (t == 0) {
    minmax[b * 4 + 0] = r0[0]; minmax[b * 4 + 1] = r1[0];
    minmax[b * 4 + 2] = r2[0]; minmax[b * 4 + 3] = r3[0];
  }
}

// ---------------------------------------------------------------------------
// Kernel 2: idx1 = argmin_p ( ||p||^2 - 2 p.q ) over 2-D, via WMMA 16x16x4 f32
// One block = 16 samples; 8 waves split the P range; LDS merge.
// Branch-free loads + tile-min gate; argmin scan only when tile improves.
// ---------------------------------------------------------------------------
__global__ void knn1_kernel(const float* __restrict__ x, const int* __restrict__ numPts,
                            const float* __restrict__ minmax,
                            const float* __restrict__ xx2, int* __restrict__ idx1) {
  int S = *numPts;
  if (S < 1) return;
  int L = (int)sqrtf((float)S);
  if (L < 1) L = 1;
  while (L * L < S) ++L;  // L = ceil(sqrt(S))
  int b  = blockIdx.x / STILES;
  int st = blockIdx.x % STILES;
  if (st * 16 >= S) return;  // uniform over the block
  int tid  = threadIdx.x;
  int lane = tid & 31;
  int wave = tid >> 5;       // 8 waves
  int n    = lane & 15;      // sample column within tile
  int half = lane >> 4;      // 0: K0/K1, 1: K2/K3
  int s  = st * 16 + n;
  int sc = (s < S) ? s : (S - 1);
  float mn0 = minmax[b * 4 + 0], mx0 = minmax[b * 4 + 1];
  float mn1 = minmax[b * 4 + 2], mx1 = minmax[b * 4 + 3];
  float Lf = (float)L;
  float qx = ((float)(sc % L) / Lf) * (mx0 - mn0) + mn0;
  float qy = ((float)(sc / L) / Lf) * (mx1 - mn1) + mn1;
  // B operand (4x16): rows K0=-2qx, K1=-2qy, K2=1, K3=0
  v2f bv;
  bv.x = (half == 0) ? (-2.0f * qx) : 1.0f;
  bv.y = (half == 0) ? (-2.0f * qy) : 0.0f;
  const float* px = x + (size_t)b * Cc * Pc;
  const float* py = px + Pc;
  const float* pr = xx2 + (size_t)b * Pc;
  // per-lane A sources, selected once (no branches in the loop):
  //   half0: K0 = x[p], K1 = y[p];  half1: K0 = ||p||^2, K1 = 0
  const float* baseA = (half == 0) ? px : pr;
  const float* baseB = py;            // loaded by all lanes; zeroed for half==1
  float kill1 = (half == 0) ? 1.0f : 0.0f;
  float best = INFINITY;
  int bestI = 0;
  int p0 = wave * 16;
  float ax = baseA[p0 + n];
  float ay = baseB[p0 + n];
  for (; p0 + 16 <= Pc; p0 += 128) {
    // prefetch next tile (clamped to a valid address on the last iteration)
    int p1 = (p0 + 144 <= Pc) ? (p0 + 128) : p0;
    float nax = baseA[p1 + n];
    float nay = baseB[p1 + n];
    v2f a;
    a.x = ax;
    a.y = ay * kill1;
    v8f c = {0.f, 0.f, 0.f, 0.f, 0.f, 0.f, 0.f, 0.f};
    c = wmma16x16x4_f32(a, bv, c);
    float tmin = min8(c);
    if (tmin < best) {  // taken only while the running min still improves
      int pib = p0 + half * 8;
#pragma unroll
      for (int v = 0; v < 8; ++v) {
        float scv = c[v];             // row M = v + 8*half, col N = lane&15
        bool bt = scv < best;         // strict < keeps lowest index on ties
        best  = bt ? scv       : best;
        bestI = bt ? (pib + v) : bestI;
      }
    }
    ax = nax;
    ay = nay;
  }
  // merge the two half-wave rows of the same sample column
  float ob = __shfl_xor(best, 16, 32);
  int   oi = __shfl_xor(bestI, 16, 32);
  bool bt = (ob < best) || (ob == best && oi < bestI);
  best  = bt ? ob : best;
  bestI = bt ? oi : bestI;
  __shared__ float sB[8][16];
  __shared__ int   sI[8][16];
  if (lane < 16) { sB[wave][lane] = best; sI[wave][lane] = bestI; }
  __syncthreads();
  if (tid < 16) {
    float bb = sB[0][tid]; int bi = sI[0][tid];
    for (int w = 1; w < 8; ++w) {
      float b2 = sB[w][tid]; int i2 = sI[w][tid];
      bool m = (b2 < bb) || (b2 == bb && i2 < bi);
      bb = m ? b2 : bb;
      bi = m ? i2 : bi;
    }
    int ss = st * 16 + tid;
    if (ss < S) idx1[(size_t)b * S + ss] = bi;
  }
}

// ---------------------------------------------------------------------------
// Kernel 3: idx2 = 8-NN in 3-D of each selected point, WMMA scoring + reg top-8
// Insertion gated by one per-tile min test; loop body branch-free otherwise.
// ---------------------------------------------------------------------------
__global__ void knn2_kernel(const float* __restrict__ x, const int* __restrict__ numPts,
                            const int* __restrict__ idx1, const float* __restrict__ xx3,
                            int* __restrict__ idx2) {
  int S = *numPts;
  if (S < 1) return;
  int b  = blockIdx.x / STILES;
  int st = blockIdx.x % STILES;
  if (st * 16 >= S) return;
  int tid  = threadIdx.x;
  int lane = tid & 31;
  int wave = tid >> 5;
  int n    = lane & 15;
  int half = lane >> 4;
  int s  = st * 16 + n;
  int sc = (s < S) ? s : (S - 1);
  const float* px = x + (size_t)b * Cc * Pc;
  const float* py = px + Pc;
  const float* pz = px + 2 * Pc;
  const float* pr = xx3 + (size_t)b * Pc;
  int qi = idx1[(size_t)b * S + sc];
  float qx = px[qi], qy = py[qi], qz = pz[qi];
  // B rows: K0=-2qx, K1=-2qy, K2=-2qz, K3=1
  v2f bv;
  bv.x = (half == 0) ? (-2.0f * qx) : (-2.0f * qz);
  bv.y = (half == 0) ? (-2.0f * qy) : 1.0f;
  // per-lane A sources: half0: (x, y), half1: (z, ||p||^2)
  const float* baseA = (half == 0) ? px : pz;
  const float* baseB = (half == 0) ? py : pr;
  float bs[8];
  int   bi[8];
#pragma unroll
  for (int j = 0; j < 8; ++j) { bs[j] = INFINITY; bi[j] = 0x7fffffff; }
  float wmax = INFINITY;  // current worst of the kept 8
  int wslot = 0;
  int p0 = wave * 16;
  float ax = baseA[p0 + n];
  float ay = baseB[p0 + n];
  for (; p0 + 16 <= Pc; p0 += 128) {
    int p1 = (p0 + 144 <= Pc) ? (p0 + 128) : p0;
    float nax = baseA[p1 + n];
    float nay = baseB[p1 + n];
    v2f a;
    a.x = ax;
    a.y = ay;
    v8f c = {0.f, 0.f, 0.f, 0.f, 0.f, 0.f, 0.f, 0.f};
    c = wmma16x16x4_f32(a, bv, c);
    float tmin = min8(c);
    if (tmin < wmax) {  // rare after warm-up: one branch per tile
      int pib = p0 + half * 8;
#pragma unroll
      for (int v = 0; v < 8; ++v) {
        float scv = c[v];
        int pi = pib + v;
        if (scv < wmax) {
#pragma unroll
          for (int j = 0; j < 8; ++j)
            if (j == wslot) { bs[j] = scv; bi[j] = pi; }
          wmax = bs[0]; wslot = 0;
#pragma unroll
          for (int j = 1; j < 8; ++j) {
            bool g = bs[j] > wmax;
            wmax  = g ? bs[j] : wmax;
            wslot = g ? j     : wslot;
          }
        }
      }
    }
    ax = nax;
    ay = nay;
  }
  // merge: 16 contributors (8 waves x 2 half-lanes) x 8 candidates per sample
  __shared__ float mS[16][128];
  __shared__ int   mI[16][128];
  int e0 = (wave * 2 + half) * 8;
#pragma unroll
  for (int j = 0; j < 8; ++j) { mS[n][e0 + j] = bs[j]; mI[n][e0 + j] = bi[j]; }
  __syncthreads();
  if (tid < 16) {
    int nn = tid;
    int ss = st * 16 + nn;
    if (ss < S) {
      for (int k = 0; k < 8; ++k) {
        float m = INFINITY; int mi = 0x7fffffff; int me = 0;
        for (int e = 0; e < 128; ++e) {
          float v = mS[nn][e];
          int ii = mI[nn][e];
          bool bt2 = (v < m) || (v == m && ii < mi);
          m  = bt2 ? v  : m;
          mi = bt2 ? ii : mi;
          me = bt2 ? e  : me;
        }
        idx2[((size_t)b * 8 + k) * S + ss] = mi;  // ascending distance = top_k order
        mS[nn][me] = INFINITY;
      }
    }
  }
}

// ---------------------------------------------------------------------------
// Kernel 4: per-sample attention network. One 128-thread block per (b, s).
// ---------------------------------------------------------------------------
__global__ void net_kernel(const float* __restrict__ x, const int* __restrict__ numPts,
                           const int* __restrict__ idx2,
                           const float* __restrict__ pos_w, const float* __restrict__ pos_b,
                           const float* __restrict__ conv_w, const float* __restrict__ conv_b,
                           const float* __restrict__ att_w1, const float* __restrict__ att_b1,
                           const float* __restrict__ att_w2, const float* __restrict__ att_b2,
                           const float* __restrict__ br1_w, const float* __restrict__ br1_b,
                           float* __restrict__ out) {
  int S = *numPts;
  int b = blockIdx.x / SMAX;
  int s = blockIdx.x % SMAX;
  if (s >= S) return;
  __shared__ float x1s[5][8];
  __shared__ float pin[10][8];
  __shared__ float feat[96][8];
  __shared__ float hbuf[32][8];
  __shared__ float attv[8];
  __shared__ float fagg[96];
  int tid = threadIdx.x;
  const float* xb = x + (size_t)b * Cc * Pc;
  if (tid < 8) {
    int k = tid;
    int p = idx2[((size_t)b * 8 + k) * S + s];
#pragma unroll
    for (int c = 0; c < 5; ++c) x1s[c][k] = xb[(size_t)c * Pc + p];
  }
  __syncthreads();
  if (tid < 8) {
    int k = tid;
    float c0 = x1s[0][0], c1 = x1s[1][0], c2 = x1s[2][0];
    float d0 = x1s[0][k] - c0, d1 = x1s[1][k] - c1, d2 = x1s[2][k] - c2;
    float t = sqrtf(d0 * d0 + d1 * d1 + d2 * d2);
    pin[0][k] = x1s[0][k]; pin[1][k] = x1s[1][k]; pin[2][k] = x1s[2][k];
    pin[3][k] = c0;        pin[4][k] = c1;        pin[5][k] = c2;
    pin[6][k] = d0;        pin[7][k] = d1;        pin[8][k] = d2;
    pin[9][k] = t;
  }
  __syncthreads();
  if (tid < 96) {  // feature rows: 0..47 conv, 48..95 pos-embed
    int o = tid;
    float acc[8];
    if (o < 48) {
#pragma unroll
      for (int k = 0; k < 8; ++k) acc[k] = conv_b[o];
#pragma unroll
      for (int i = 0; i < 4; ++i) {
        float w = conv_w[o * 4 + i];
#pragma unroll
        for (int k = 0; k < 8; ++k) acc[k] = fmaf(w, x1s[i][k], acc[k]);
      }
    } else {
      int oo = o - 48;
#pragma unroll
      for (int k = 0; k < 8; ++k) acc[k] = pos_b[oo];
#pragma unroll
      for (int i = 0; i < 10; ++i) {
        float w = pos_w[oo * 10 + i];
#pragma unroll
        for (int k = 0; k < 8; ++k) acc[k] = fmaf(w, pin[i][k], acc[k]);
      }
    }
#pragma unroll
    for (int k = 0; k < 8; ++k) feat[o][k] = acc[k];
  }
  __syncthreads();
  if (tid < 32) {  // h = leaky_relu(att_w1 @ feature + b1)
    int j = tid;
    float acc[8];
#pragma unroll
    for (int k = 0; k < 8; ++k) acc[k] = att_b1[j];
    for (int i = 0; i < 96; ++i) {
      float w = att_w1[j * 96 + i];
#pragma unroll
      for (int k = 0; k < 8; ++k) acc[k] = fmaf(w, feat[i][k], acc[k]);
    }
#pragma unroll
    for (int k = 0; k < 8; ++k) {
      float v = acc[k];
      hbuf[j][k] = (v >= 0.f) ? v : 0.2f * v;
    }
  }
  __syncthreads();
  if (tid < 8) {
    int k = tid;
    float acc = att_b2[0];
#pragma unroll
    for (int j = 0; j < 32; ++j) acc = fmaf(att_w2[j], hbuf[j][k], acc);
    attv[k] = acc;
  }
  __syncthreads();
  if (tid == 0) {  // softmax over the 8 neighbors
    float mx = attv[0];
#pragma unroll
    for (int k = 1; k < 8; ++k) mx = fmaxf(mx, attv[k]);
    float e[8];
    float sum = 0.f;
#pragma unroll
    for (int k = 0; k < 8; ++k) { e[k] = expf(attv[k] - mx); sum += e[k]; }
    float inv = 1.f / sum;
#pragma unroll
    for (int k = 0; k < 8; ++k) attv[k] = e[k] * inv;
  }
  __syncthreads();
  if (tid < 96) {
    float acc = 0.f;
#pragma unroll
    for (int k = 0; k < 8; ++k) acc = fmaf(feat[tid][k], attv[k], acc);
    fagg[tid] = acc;
  }
  __syncthreads();
  size_t ob = (size_t)b * 101 * S;
  if (tid < 96) {
    int o = tid;
    float acc = br1_b[o];
    for (int i = 0; i < 96; ++i) acc = fmaf(br1_w[o * 96 + i], fagg[i], acc);
    out[ob + (size_t)(5 + o) * S + s] = acc;
  }
  if (tid < 5) {
    out[ob + (size_t)tid * S + s] = x1s[tid][0];
  }
}

// ---------------------------------------------------------------------------
extern "C" void kernel_launch(void* const* d_in, const int* in_sizes, int n_in,
                              void* d_out, int out_size, void* d_ws, size_t ws_size,
                              hipStream_t stream) {
  (void)in_sizes; (void)n_in; (void)out_size; (void)ws_size;
  const float* x      = (const float*)d_in[0];
  const int* numPts   = (const int*)d_in[1];
  const float* pos_w  = (const float*)d_in[2];
  const float* pos_b  = (const float*)d_in[3];
  const float* conv_w = (const float*)d_in[4];
  const float* conv_b = (const float*)d_in[5];
  const float* att_w1 = (const float*)d_in[6];
  const float* att_b1 = (const float*)d_in[7];
  const float* att_w2 = (const float*)d_in[8];
  const float* att_b2 = (const float*)d_in[9];
  const float* br1_w  = (const float*)d_in[10];
  const float* br1_b  = (const float*)d_in[11];
  float* out = (float*)d_out;

  char* ws = (char*)d_ws;
  float* minmax = (float*)ws;                       // 16 floats (padded to 256B)
  float* xx2    = (float*)(ws + 256);               // B*P floats
  float* xx3    = xx2 + (size_t)Bc * Pc;            // B*P floats
  int*   idx1   = (int*)(xx3 + (size_t)Bc * Pc);    // B*SMAX ints
  int*   idx2   = idx1 + (size_t)Bc * SMAX;         // B*8*SMAX ints

  prep_kernel<<<Bc, 256, 0, stream>>>(x, minmax, xx2, xx3);
  knn1_kernel<<<Bc * STILES, 256, 0, stream>>>(x, numPts, minmax, xx2, idx1);
  knn2_kernel<<<Bc * STILES, 256, 0, stream>>>(x, numPts, idx1, xx3, idx2);
  net_kernel<<<Bc * SMAX, 128, 0, stream>>>(x, numPts, idx2, pos_w, pos_b,
                                            conv_w, conv_b, att_w1, att_b1,
                                            att_w2, att_b2, br1_w, br1_b, out);
}